// GNNEncoder_78804059947264
// MI455X (gfx1250) — compile-verified
//
#include <hip/hip_runtime.h>

typedef float v2f __attribute__((ext_vector_type(2)));
typedef float v4f __attribute__((ext_vector_type(4)));
typedef float v8f __attribute__((ext_vector_type(8)));

#define N_NODES 100000
#define F 128
#define EDGES 1600000
#define ROWS_PB 80      // rows per block: 100000 / 80 = 1250 blocks exactly
#define LDSP 132        // padded LDS row stride (floats): conflict-free fragment reads

// ---------------- zero workspace (msg_sum + cnt) ----------------
__global__ void zero_ws(float* __restrict__ p, int n4) {
  int i = blockIdx.x * blockDim.x + threadIdx.x;
  int stride = gridDim.x * blockDim.x;
  v4f z = {0.0f, 0.0f, 0.0f, 0.0f};
  for (; i < n4; i += stride) ((v4f*)p)[i] = z;
}

// ---------------- edge scatter: one wave32 per edge ----------------
// lane l owns floats [4l, 4l+4) of the 128-wide feature row.
__global__ __launch_bounds__(256) void scatter_edges(
    const int* __restrict__ ei, const float* __restrict__ x,
    float* __restrict__ msum, float* __restrict__ cnt) {
  int e = blockIdx.x * 8 + (threadIdx.x >> 5);
  int lane = threadIdx.x & 31;
  int src = ei[e];
  int dst = ei[EDGES + e];
  v4f xv = *(const v4f*)(x + (size_t)src * F + lane * 4);
  float* base = msum + (size_t)dst * F + lane * 4;
  unsafeAtomicAdd(base + 0, xv.x);
  unsafeAtomicAdd(base + 1, xv.y);
  unsafeAtomicAdd(base + 2, xv.z);
  unsafeAtomicAdd(base + 3, xv.w);
  if (lane == 0) unsafeAtomicAdd(cnt + dst, 1.0f);
}

// ---------------- fused mean-normalize + dual GEMM + bias + ReLU ----------------
// out[i,j] = relu( sum_k mean[i,k]*Wl[j,k] + bl[j] + sum_k x[i,k]*Wr[j,k] )
// Block: 80 rows, 8 waves; wave w -> columns [16w,16w+16), 5 stacked 16x16 tiles.
__global__ __launch_bounds__(256) void sage_gemm(
    const float* __restrict__ msum, const float* __restrict__ cnt,
    const float* __restrict__ x,
    const float* __restrict__ Wl, const float* __restrict__ bl,
    const float* __restrict__ Wr, float* __restrict__ out) {
  __shared__ float ldsM[ROWS_PB * LDSP];
  __shared__ float ldsX[ROWS_PB * LDSP];

  const int t  = threadIdx.x;
  const int i0 = blockIdx.x * ROWS_PB;

  // Stage A tiles into LDS (coalesced float4), normalizing msg_sum by count.
  for (int q = t; q < ROWS_PB * (F / 4); q += 256) {
    int row = q >> 5;            // 32 float4 per 128-wide row
    int c4  = q & 31;
    int g   = i0 + row;
    float rc = 1.0f / fmaxf(cnt[g], 1.0f);
    v4f m = *(const v4f*)(msum + (size_t)g * F + c4 * 4);
    m.x *= rc; m.y *= rc; m.z *= rc; m.w *= rc;
    *(v4f*)(ldsM + row * LDSP + c4 * 4) = m;
    *(v4f*)(ldsX + row * LDSP + c4 * 4) = *(const v4f*)(x + (size_t)g * F + c4 * 4);
  }
  __syncthreads();

  const int wave = t >> 5;
  const int lane = t & 31;
  const int j0 = wave * 16;
  const int nh = lane & 15;           // A-row / B-col within tile
  const int kh = (lane >> 4) << 1;    // k sub-offset: 0 or 2

  float bias = bl[j0 + nh];
  v8f acc[5];
#pragma unroll
  for (int s = 0; s < 5; ++s) {
#pragma unroll
    for (int r = 0; r < 8; ++r) acc[s][r] = bias;   // bias depends only on column
  }

  for (int kb = 0; kb < F; kb += 4) {
    // B fragments from W (row j = output col, contiguous k): reused across 5 subtiles
    v2f bL = *(const v2f*)(Wl + (size_t)(j0 + nh) * F + kb + kh);
    v2f bR = *(const v2f*)(Wr + (size_t)(j0 + nh) * F + kb + kh);
#pragma unroll
    for (int s = 0; s < 5; ++s) {
      v2f aM = *(const v2f*)(ldsM + (s * 16 + nh) * LDSP + kb + kh);
      v2f aX = *(const v2f*)(ldsX + (s * 16 + nh) * LDSP + kb + kh);
      acc[s] = __builtin_amdgcn_wmma_f32_16x16x4_f32(
          false, aM, false, bL, (short)0, acc[s], false, false);
      acc[s] = __builtin_amdgcn_wmma_f32_16x16x4_f32(
          false, aX, false, bR, (short)0, acc[s], false, false);
    }
  }

  // Store with ReLU. C/D layout: VGPR r -> row (r + 8*(lane>=16)), col = lane&15.
#pragma unroll
  for (int s = 0; s < 5; ++s) {
    int rbase = i0 + s * 16 + ((lane >> 4) << 3);
#pragma unroll
    for (int r = 0; r < 8; ++r) {
      out[(size_t)(rbase + r) * F + j0 + nh] = fmaxf(acc[s][r], 0.0f);
    }
  }
}

extern "C" void kernel_launch(void* const* d_in, const int* in_sizes, int n_in,
                              void* d_out, int out_size, void* d_ws, size_t ws_size,
                              hipStream_t stream) {
  const float* x  = (const float*)d_in[0];
  const int*   ei = (const int*)d_in[1];     // edge_index (JAX x64 off -> int32)
  const float* Wl = (const float*)d_in[2];
  const float* bl = (const float*)d_in[3];
  const float* Wr = (const float*)d_in[4];
  float* out = (float*)d_out;

  float* msum = (float*)d_ws;                       // N*128 floats
  float* cnt  = msum + (size_t)N_NODES * F;         // N floats

  int n4 = (N_NODES * (F + 1)) / 4;                 // zero msum + cnt together
  zero_ws<<<2048, 256, 0, stream>>>(msum, n4);
  scatter_edges<<<EDGES / 8, 256, 0, stream>>>(ei, x, msum, cnt);
  sage_gemm<<<N_NODES / ROWS_PB, 256, 0, stream>>>(msum, cnt, x, Wl, bl, Wr, out);
}